// PixelBitFiLayer_72138270703956
// MI455X (gfx1250) — compile-verified
//
#include <hip/hip_runtime.h>
#include <hip/hip_bf16.h>
#include <stdint.h>

// Problem constants (match reference: B=64, H=512, W=512, C=3, NBITS=8, N_FAULTS=4096)
#define DIM_B   64
#define DIM_H   512
#define DIM_W   512
#define DIM_C   3
#define HWC     (DIM_H * DIM_W * DIM_C)     // 786432 int32 elements (3 MB)
#define NFAULTS 4096

// Native clang vector -> legal for __builtin_nontemporal_* and lowers to B128.
typedef int vint4 __attribute__((ext_vector_type(4)));

// ---------------------------------------------------------------------------
// Kernel 1: zero the fault mask (d_ws is NOT re-poisoned between replays, so
// we must re-zero every launch). 786432 ints = 196608 vint4 -> 768 blocks.
// ---------------------------------------------------------------------------
__global__ __launch_bounds__(256) void fi_zero_mask(vint4* __restrict__ mask) {
    unsigned i = blockIdx.x * 256u + threadIdx.x;
    mask[i] = (vint4){0, 0, 0, 0};
}

// ---------------------------------------------------------------------------
// Kernel 2: scatter the 4096 bit faults into the mask. Faults are unique per
// (y,x,c,bit), so atomicOr == the reference's scatter-add. Order-independent
// -> deterministic across replays.
// ---------------------------------------------------------------------------
__global__ __launch_bounds__(256) void fi_scatter(const int* __restrict__ faults,
                                                  int* __restrict__ mask) {
    unsigned f = blockIdx.x * 256u + threadIdx.x;
    if (f < NFAULTS) {
        int y   = faults[4 * f + 0];
        int x   = faults[4 * f + 1];
        int c   = faults[4 * f + 2];
        int bit = faults[4 * f + 3];
        int idx = (y * DIM_W + x) * DIM_C + c;
        atomicOr(&mask[idx], 1 << bit);
    }
}

// ---------------------------------------------------------------------------
// Kernel 3: streaming XOR.  grid = (HWC/1024, B).  Each 256-thread block
// (8 wave32) owns a contiguous 4 KB image chunk of batch image blockIdx.y.
// The matching 4 KB mask chunk is staged into LDS with the CDNA5 async
// memory path (global_load_async_to_lds_b128, tracked by ASYNCcnt).  Each
// thread consumes exactly the LDS slot its own wave fetched, so a per-wave
// s_wait_asynccnt 0 is sufficient (no workgroup barrier required).
// Image traffic is one-touch (403 MB total) -> non-temporal TH hints keep it
// from evicting the hot 3 MB mask out of the 192 MB L2.
// ---------------------------------------------------------------------------
__global__ __launch_bounds__(256) void fi_xor_stream(const vint4* __restrict__ img,
                                                     const vint4* __restrict__ mask,
                                                     vint4* __restrict__ out) {
    __shared__ vint4 smask[256];

    const unsigned t    = threadIdx.x;
    const unsigned mvec = blockIdx.x * 256u + t;                 // vint4 index into mask
    const unsigned long long gvec =
        (unsigned long long)blockIdx.y * (unsigned long long)(HWC / 4) + mvec;

    // LDS byte offset of this thread's staging slot (flat-addr low 32 bits ==
    // LDS offset per the CDNA5 aperture mapping).
    unsigned lds_off  = (unsigned)(uintptr_t)(&smask[t]);
    unsigned gbyte    = mvec * 16u;                              // byte offset in mask
    unsigned long long mbase = (unsigned long long)(uintptr_t)mask;

    // CDNA5 async copy: copies 16 bytes/lane from mask at byte offset gbyte to the LDS address in lds_off.
    asm volatile("global_load_async_to_lds_b128 %0, %1, %2"
                 :
                 : "v"(lds_off), "v"(gbyte), "s"(mbase)
                 : "memory");

    // Overlap: pull the one-touch image vector while the async copy runs.
    vint4 v = __builtin_nontemporal_load(&img[gvec]);

    // Drain this wave's ASYNCcnt before touching LDS.
    asm volatile("s_wait_asynccnt 0" ::: "memory");
    vint4 m = smask[t];

    vint4 r = v ^ m;
    __builtin_nontemporal_store(r, &out[gvec]);
}

// ---------------------------------------------------------------------------
// Fallback path (used only if ws_size < 3 MB): stream-copy image -> out, then
// XOR the faults directly into every batch image (4096 x 64 = 256K atomics,
// ~1 MB of L2 traffic, negligible vs the 403 MB stream).
// ---------------------------------------------------------------------------
__global__ __launch_bounds__(256) void fi_copy_stream(const vint4* __restrict__ img,
                                                      vint4* __restrict__ out) {
    size_t i = (size_t)blockIdx.x * 256u + threadIdx.x;
    __builtin_nontemporal_store(__builtin_nontemporal_load(&img[i]), &out[i]);
}

__global__ __launch_bounds__(256) void fi_fault_xor(const int* __restrict__ faults,
                                                    int* __restrict__ out) {
    unsigned f = blockIdx.x * 256u + threadIdx.x;
    if (f < NFAULTS) {
        int y   = faults[4 * f + 0];
        int x   = faults[4 * f + 1];
        int c   = faults[4 * f + 2];
        int bit = faults[4 * f + 3];
        size_t idx = (size_t)blockIdx.y * HWC + (size_t)(y * DIM_W + x) * DIM_C + c;
        atomicXor(&out[idx], 1 << bit);
    }
}

// ---------------------------------------------------------------------------
extern "C" void kernel_launch(void* const* d_in, const int* in_sizes, int n_in,
                              void* d_out, int out_size, void* d_ws, size_t ws_size,
                              hipStream_t stream) {
    const int* img    = (const int*)d_in[0];   // (B,H,W,C) int32
    const int* faults = (const int*)d_in[1];   // (NFAULTS,4) int32
    int*       out    = (int*)d_out;

    const size_t mask_bytes = (size_t)HWC * sizeof(int);

    if (ws_size >= mask_bytes) {
        int* mask = (int*)d_ws;

        // 1) zero mask (must happen every call: ws state persists across replays)
        fi_zero_mask<<<HWC / 4 / 256, 256, 0, stream>>>((vint4*)mask);

        // 2) scatter faults
        fi_scatter<<<(NFAULTS + 255) / 256, 256, 0, stream>>>(faults, mask);

        // 3) broadcast XOR over the batch; mask chunk staged via async-to-LDS
        dim3 grid(HWC / 1024, DIM_B);          // (768, 64)
        fi_xor_stream<<<grid, 256, 0, stream>>>((const vint4*)img,
                                                (const vint4*)mask,
                                                (vint4*)out);
    } else {
        // Fallback: copy stream then per-batch fault XOR (identical semantics).
        const unsigned nvec = (unsigned)((size_t)DIM_B * HWC / 4);  // 12582912
        fi_copy_stream<<<nvec / 256, 256, 0, stream>>>((const vint4*)img, (vint4*)out);
        dim3 fgrid((NFAULTS + 255) / 256, DIM_B);
        fi_fault_xor<<<fgrid, 256, 0, stream>>>(faults, out);
    }
}